// LinearDynamicModel_84301618085952
// MI455X (gfx1250) — compile-verified
//
#include <hip/hip_runtime.h>
#include <hip/hip_bf16.h>

#define L_   8
#define NB_  8192
#define D_   1024

typedef float v2f __attribute__((ext_vector_type(2)));
typedef float v8f __attribute__((ext_vector_type(8)));

// ---------------------------------------------------------------------------
// Kernel 1: spectral norm of (M_l + diag(d_l)) via power iteration on M~^T M~.
// One 1024-thread block per matrix (16 matrices: 8 W-lags then 8 B-lags).
// Writes scale_l = (sigma > 2) ? 2/sigma : 1 to scales[m].
// ---------------------------------------------------------------------------
__device__ inline float blk_reduce_sum(float val, float* red) {
#pragma unroll
  for (int off = 16; off > 0; off >>= 1) val += __shfl_down(val, off);
  const int lane = threadIdx.x & 31;
  const int wv   = threadIdx.x >> 5;
  if (lane == 0) red[wv] = val;
  __syncthreads();
  if (wv == 0) {
    float r = red[lane];
#pragma unroll
    for (int off = 16; off > 0; off >>= 1) r += __shfl_down(r, off);
    if (lane == 0) red[0] = r;
  }
  __syncthreads();
  float outv = red[0];
  __syncthreads();
  return outv;
}

__global__ __launch_bounds__(1024) void spectral_norm_kernel(
    const float* __restrict__ W, const float* __restrict__ alpha,
    const float* __restrict__ Bm, const float* __restrict__ beta,
    float* __restrict__ scales)
{
  __shared__ float v[D_];
  __shared__ float u[D_];
  __shared__ float red[32];

  const int  m   = blockIdx.x;       // 0..15
  const int  l   = m & 7;
  const bool isW = (m < 8);
  const float* M  = (isW ? W : Bm) + (size_t)l * D_ * D_;
  const float* dv = (isW ? alpha : beta) + (size_t)l * D_;

  const int t    = threadIdx.x;
  const int lane = t & 31;
  const int wv   = t >> 5;

  v[t] = 0.03125f;                   // unit-norm start (1024 * (1/32)^2 == 1)
  __syncthreads();

  float sigma2 = 0.f;
#pragma unroll 1
  for (int it = 0; it < 10; ++it) {
    // u = (M + diag(dv)) v : each wave owns 32 rows, lanes split columns.
#pragma unroll 1
    for (int rr = 0; rr < 32; ++rr) {
      const int r = wv * 32 + rr;
      const float* Mr = M + (size_t)r * D_;
      float p = 0.f;
#pragma unroll 4
      for (int j = lane; j < D_; j += 32) p += Mr[j] * v[j];
#pragma unroll
      for (int off = 16; off > 0; off >>= 1) p += __shfl_down(p, off);
      if (lane == 0) u[r] = p + dv[r] * v[r];
    }
    __syncthreads();

    // Rayleigh estimate: ||u||^2 (v was unit norm).
    sigma2 = blk_reduce_sum(u[t] * u[t], red);

    // z = (M + diag(dv))^T u : thread t owns column t (coalesced per i-row).
    float z = dv[t] * u[t];
#pragma unroll 4
    for (int i = 0; i < D_; ++i) z += M[(size_t)i * D_ + t] * u[i];

    float nz = blk_reduce_sum(z * z, red);
    v[t] = z * rsqrtf(fmaxf(nz, 1e-30f));
    __syncthreads();
  }

  if (t == 0) {
    float s = sqrtf(sigma2);
    scales[m] = (s > 2.0f) ? (2.0f / s) : 1.0f;
  }
}

// ---------------------------------------------------------------------------
// Kernel 2: fused dual GEMM with WMMA f32 16x16x4, software-pipelined.
//   out[b,i] = sum_l s_l*(W_l+diag a_l)·x_l + sum_l t_l*(B_l+diag b_l)·u_l + V
// Block tile: 128 batch x 64 feat, 8 waves. K-tile 32, flattened into 512
// stages (l, src, j0). Double-buffered LDS: A tiles (X/U) via async
// global->LDS b128 prefetched one stage ahead; raw split barriers with
// manual s_wait_asynccnt<=4 keep the next tile's loads in flight across the
// barrier (builtin __syncthreads would drain ASYNCcnt to 0 and kill overlap).
// W/B tiles via normal loads with fused diag-add + spectral scale.
// ---------------------------------------------------------------------------
#define KT  32
#define LDA 36   // padded stride: 16-lane groups hit 16 distinct banks
#define LDW 36
#define NS  (L_ * 2 * (D_ / KT))   // 512 pipeline stages

__device__ __forceinline__ void async_copy_b128(unsigned lds_byte_off,
                                                const void* gptr) {
  asm volatile("global_load_async_to_lds_b128 %0, %1, off"
               :: "v"(lds_byte_off), "v"(gptr)
               : "memory");
}
__device__ __forceinline__ void wait_async_le4() {
  asm volatile("s_wait_asynccnt 4" ::: "memory");
}
__device__ __forceinline__ void wait_async_le0() {
  asm volatile("s_wait_asynccnt 0" ::: "memory");
}
__device__ __forceinline__ void wait_ds0() {
  asm volatile("s_wait_dscnt 0" ::: "memory");
}
__device__ __forceinline__ void wg_barrier() {
  asm volatile("s_barrier_signal -1\n\ts_barrier_wait -1" ::: "memory");
}

// Issue the 4 async b128 loads for stage st's A tile (128 x 32 floats).
__device__ __forceinline__ void issue_A_tile(const float* __restrict__ Xg,
                                             const float* __restrict__ Ug,
                                             int st, int tid, int b0,
                                             float* dstBase) {
  const int l   = st >> 6;
  const int src = (st >> 5) & 1;
  const int j0  = (st & 31) * KT;
  const float* Ap = src ? Ug : Xg;
#pragma unroll
  for (int rep = 0; rep < 4; ++rep) {
    const int q    = tid + rep * 256;
    const int mrow = q >> 3;
    const int kq   = q & 7;
    const float* g = Ap + ((size_t)l * NB_ + (size_t)(b0 + mrow)) * D_
                        + j0 + kq * 4;
    async_copy_b128((unsigned)(size_t)&dstBase[mrow * LDA + kq * 4], g);
  }
}

__global__ __launch_bounds__(256) void fused_gemm_kernel(
    const float* __restrict__ X, const float* __restrict__ U,
    const float* __restrict__ W, const float* __restrict__ alpha,
    const float* __restrict__ Bm, const float* __restrict__ beta,
    const float* __restrict__ V, const float* __restrict__ scales,
    float* __restrict__ out)
{
  __shared__ __align__(16) float At[2][128 * LDA];   // 2 x 18 KB
  __shared__ __align__(16) float Wt[2][64 * LDW];    // 2 x  9 KB

  const int tid  = threadIdx.x;
  const int lane = tid & 31;
  const int w    = tid >> 5;         // wave 0..7
  const int half = lane >> 4;        // 0|1 (lane group)
  const int l15  = lane & 15;
  const int b0   = blockIdx.x * 128; // batch tile
  const int i0   = blockIdx.y * 64;  // feature tile

  v8f acc[4] = {};                   // one 16x16 f32 tile per ntile

  // ---- prologue: prefetch stage 0's A tile
  issue_A_tile(X, U, 0, tid, b0, At[0]);

#pragma unroll 1
  for (int st = 0; st < NS; ++st) {
    const int buf = st & 1;
    const int l   = st >> 6;
    const int src = (st >> 5) & 1;
    const int j0  = (st & 31) * KT;
    const float* Mp  = src ? Bm   : W;
    const float* dvp = src ? beta : alpha;
    const float  s   = scales[src * 8 + l];

    // ---- stage W tile (64 x 32) with fused (+diag, *scale) -> Wt[buf]
    {
      float* Wb = Wt[buf];
#pragma unroll
      for (int rep = 0; rep < 2; ++rep) {
        const int q    = tid + rep * 256;
        const int irow = q >> 3;
        const int kq   = q & 7;
        const int ig   = i0 + irow;
        const int jg   = j0 + kq * 4;
        float4 wv4 = *(const float4*)(Mp + ((size_t)l * D_ + ig) * D_ + jg);
        const float dval = dvp[l * D_ + ig];
        if (ig == jg + 0) wv4.x += dval;
        if (ig == jg + 1) wv4.y += dval;
        if (ig == jg + 2) wv4.z += dval;
        if (ig == jg + 3) wv4.w += dval;
        wv4.x *= s; wv4.y *= s; wv4.z *= s; wv4.w *= s;
        *(float4*)&Wb[irow * LDW + kq * 4] = wv4;
      }
    }

    // ---- prefetch next stage's A tile into the other buffer, then wait for
    //      THIS stage only (async loads complete in order: <=4 outstanding
    //      leaves exactly the next tile's 4 loads in flight).
    if (st + 1 < NS) {
      issue_A_tile(X, U, st + 1, tid, b0, At[buf ^ 1]);
      wait_async_le4();
    } else {
      wait_async_le0();
    }
    wait_ds0();          // own Wt stores visible
    wg_barrier();        // split barrier: does NOT drain the prefetch

    // ---- 8 K-steps of WMMA f32 16x16x4: batch all LDS fragment loads per
    //      kk before issuing the 4 WMMAs (one dscnt wait instead of four).
    {
      const float* Ab = At[buf];
      const float* Wb = Wt[buf];
#pragma unroll
      for (int kk = 0; kk < KT / 4; ++kk) {
        const int kbase = kk * 4 + half * 2; // A/B: VGPR0=K{0,2}, VGPR1=K{1,3}
        v2f a;
        a.x = Ab[(w * 16 + l15) * LDA + kbase];
        a.y = Ab[(w * 16 + l15) * LDA + kbase + 1];
        v2f bf[4];
#pragma unroll
        for (int nt = 0; nt < 4; ++nt) {
          bf[nt].x = Wb[(nt * 16 + l15) * LDW + kbase];
          bf[nt].y = Wb[(nt * 16 + l15) * LDW + kbase + 1];
        }
#pragma unroll
        for (int nt = 0; nt < 4; ++nt) {
          acc[nt] = __builtin_amdgcn_wmma_f32_16x16x4_f32(
              false, a, false, bf[nt], (short)0, acc[nt], false, false);
        }
      }
    }
    wg_barrier();        // all waves done reading before next stage overwrites
  }

  // ---- epilogue: + V, scatter per C/D layout (VGPR r: M=r / r+8)
#pragma unroll
  for (int nt = 0; nt < 4; ++nt) {
    const int col = i0 + nt * 16 + l15;
    const float vb = V[col];
#pragma unroll
    for (int r = 0; r < 8; ++r) {
      const int row = b0 + w * 16 + r + half * 8;
      out[(size_t)row * D_ + col] = acc[nt][r] + vb;
    }
  }
}

// ---------------------------------------------------------------------------
extern "C" void kernel_launch(void* const* d_in, const int* in_sizes, int n_in,
                              void* d_out, int out_size, void* d_ws, size_t ws_size,
                              hipStream_t stream) {
  (void)in_sizes; (void)n_in; (void)out_size; (void)ws_size;
  const float* X     = (const float*)d_in[0];
  const float* U     = (const float*)d_in[1];
  const float* W     = (const float*)d_in[2];
  const float* alpha = (const float*)d_in[3];
  const float* Bm    = (const float*)d_in[4];
  const float* beta  = (const float*)d_in[5];
  const float* V     = (const float*)d_in[6];
  float* out    = (float*)d_out;
  float* scales = (float*)d_ws;   // 16 floats

  hipLaunchKernelGGL(spectral_norm_kernel, dim3(16), dim3(1024), 0, stream,
                     W, alpha, Bm, beta, scales);
  hipLaunchKernelGGL(fused_gemm_kernel, dim3(NB_ / 128, D_ / 64), dim3(256), 0, stream,
                     X, U, W, alpha, Bm, beta, V, scales, out);
}